// GraphConvolution_14474039787903
// MI455X (gfx1250) — compile-verified
//
#include <hip/hip_runtime.h>

typedef __attribute__((ext_vector_type(2))) float v2f;
typedef __attribute__((ext_vector_type(4))) float v4f;
typedef __attribute__((ext_vector_type(8))) float v8f;

#define K_DIM   256
#define N_DIM   128
#define LDS_STR 260   // 256 + 4 pad: bank step 4 -> conflict-free 16-lane column reads

// ---------------------------------------------------------------------------
// Kernel 1: pre_sup[M,128] = x[M,256] @ W[256,128]  (fp32 WMMA 16x16x4)
// One workgroup = 128 threads = 4 waves. Block owns a 16-row strip of x.
// Each wave owns two 16-wide N tiles (4 waves * 2 * 16 = 128 = N_DIM).
// ---------------------------------------------------------------------------
__global__ __launch_bounds__(128)
void gcn_gemm_wmma_f32(const float* __restrict__ x, const float* __restrict__ W,
                       float* __restrict__ pre, int n_nodes) {
    __shared__ float xs[16 * LDS_STR];

    const int m0  = blockIdx.x * 16;
    const int tid = threadIdx.x;

    // Cooperative, coalesced load of the 16x256 x tile into LDS (float4).
    // 16 rows * 64 float4-chunks = 1024 chunks / 128 threads = 8 iterations.
    for (int i = tid; i < 16 * 64; i += 128) {
        const int r  = i >> 6;
        const int c4 = i & 63;
        const int gr = m0 + r;
        v4f v = {0.0f, 0.0f, 0.0f, 0.0f};
        if (gr < n_nodes)
            v = *(const v4f*)(x + (size_t)gr * K_DIM + (size_t)c4 * 4);
        *(v4f*)(&xs[r * LDS_STR + c4 * 4]) = v;
    }
    __syncthreads();

    const int wave = tid >> 5;
    const int lane = tid & 31;
    const int half = lane >> 4;   // 0: lanes 0-15, 1: lanes 16-31
    const int l16  = lane & 15;

    const int n0 = (wave * 2) * 16;
    const int n1 = n0 + 16;

    v8f acc0 = {};
    v8f acc1 = {};

    // K loop: 64 WMMA steps of K=4. A frag shared by both N tiles.
    for (int k = 0; k < K_DIM; k += 4) {
        // A 16x4 f32: VGPR j = K (k + 2*half + j), row M = l16
        v2f a = *(const v2f*)(&xs[l16 * LDS_STR + k + 2 * half]);

        // B 4x16 f32: VGPR j = K (k + 2*half + j), col N = l16 (+ tile base)
        const float* wp = W + (size_t)(k + 2 * half) * N_DIM;
        v2f b0, b1;
        b0.x = wp[n0 + l16];
        b0.y = wp[N_DIM + n0 + l16];
        b1.x = wp[n1 + l16];
        b1.y = wp[N_DIM + n1 + l16];

        acc0 = __builtin_amdgcn_wmma_f32_16x16x4_f32(false, a, false, b0,
                                                     (short)0, acc0, false, false);
        acc1 = __builtin_amdgcn_wmma_f32_16x16x4_f32(false, a, false, b1,
                                                     (short)0, acc1, false, false);
    }

    // D 16x16 f32: VGPR r holds M = r + 8*half, N = l16
    #pragma unroll
    for (int r = 0; r < 8; ++r) {
        const int gm = m0 + r + 8 * half;
        if (gm < n_nodes) {
            pre[(size_t)gm * N_DIM + n0 + l16] = acc0[r];
            pre[(size_t)gm * N_DIM + n1 + l16] = acc1[r];
        }
    }
}

// ---------------------------------------------------------------------------
// Kernel 2: out[n][j] = b[j]  (bias as the accumulator init)
// ---------------------------------------------------------------------------
__global__ __launch_bounds__(256)
void gcn_init_bias(float* __restrict__ out, const float* __restrict__ b, int total) {
    const int i = blockIdx.x * 256 + threadIdx.x;
    if (i < total) out[i] = b[i & (N_DIM - 1)];
}

// ---------------------------------------------------------------------------
// Kernel 3: scatter-add.  One wave32 per edge; lane owns 4 contiguous floats.
// pre_sup (25.6 MB) and out (25.6 MB) are L2-resident; atomics are native
// global_atomic_add_f32 (unsafeAtomicAdd), not CAS loops.
// ---------------------------------------------------------------------------
__global__ __launch_bounds__(256)
void gcn_spmm_scatter(const int* __restrict__ rows, const int* __restrict__ cols,
                      const float* __restrict__ vals, const float* __restrict__ pre,
                      float* __restrict__ out, int n_edges) {
    const int lane = threadIdx.x & 31;
    // Edge id is wave-uniform: force into an SGPR so row/col/val become scalar loads.
    int e = __builtin_amdgcn_readfirstlane(blockIdx.x * 8 + (threadIdx.x >> 5));
    if (e >= n_edges) return;

    const int   r = rows[e];
    const int   c = cols[e];
    const float v = vals[e];

    const v4f p = *(const v4f*)(pre + (size_t)c * N_DIM + lane * 4);
    float* o = out + (size_t)r * N_DIM + lane * 4;

    unsafeAtomicAdd(o + 0, v * p.x);
    unsafeAtomicAdd(o + 1, v * p.y);
    unsafeAtomicAdd(o + 2, v * p.z);
    unsafeAtomicAdd(o + 3, v * p.w);
}

// ---------------------------------------------------------------------------
// Kernel 4: in-place ReLU
// ---------------------------------------------------------------------------
__global__ __launch_bounds__(256)
void gcn_relu(float* __restrict__ out, int total) {
    const int i = blockIdx.x * 256 + threadIdx.x;
    if (i < total) out[i] = fmaxf(out[i], 0.0f);
}

// ---------------------------------------------------------------------------
// Launch. Inputs (setup_inputs order): x, edge_rows, edge_cols, edge_vals, W, b
// ---------------------------------------------------------------------------
extern "C" void kernel_launch(void* const* d_in, const int* in_sizes, int n_in,
                              void* d_out, int out_size, void* d_ws, size_t ws_size,
                              hipStream_t stream) {
    const float* x    = (const float*)d_in[0];
    const int*   rows = (const int*)  d_in[1];
    const int*   cols = (const int*)  d_in[2];
    const float* vals = (const float*)d_in[3];
    const float* W    = (const float*)d_in[4];
    const float* b    = (const float*)d_in[5];
    float*       out  = (float*)d_out;

    const int n_nodes = in_sizes[0] / K_DIM;   // 50000
    const int n_edges = in_sizes[1];           // 1.6M
    const int total   = n_nodes * N_DIM;       // 6.4M

    float* pre = (float*)d_ws;                 // 25.6 MB scratch: pre_sup

    // 1) pre_sup = x @ W via V_WMMA_F32_16X16X4_F32
    {
        dim3 grid((n_nodes + 15) / 16);
        gcn_gemm_wmma_f32<<<grid, 128, 0, stream>>>(x, W, pre, n_nodes);
    }
    // 2) out = bias (broadcast)
    {
        dim3 grid((total + 255) / 256);
        gcn_init_bias<<<grid, 256, 0, stream>>>(out, b, total);
    }
    // 3) out[row] += val * pre_sup[col]  (one wave per edge, 8 edges/block)
    {
        dim3 grid((n_edges + 7) / 8);
        gcn_spmm_scatter<<<grid, 256, 0, stream>>>(rows, cols, vals, pre, out, n_edges);
    }
    // 4) ReLU
    {
        dim3 grid((total + 255) / 256);
        gcn_relu<<<grid, 256, 0, stream>>>(out, total);
    }
}